// DifferentiableMDSModule_86921548136926
// MI455X (gfx1250) — compile-verified
//
#include <hip/hip_runtime.h>
#include <hip/hip_bf16.h>
#include <math.h>

// ---------------------------------------------------------------------------
// Batched MDS: B=64, L=512, top-3 eigenpairs of Bmat = -0.5*J*d2*J + jitter*I
//   K0: zero per-batch sigma (Gershgorin) accumulators
//   K1: per-batch masked row sums r_i, total S, count n      (streaming)
//   K2: closed-form double centering -> Bmat rows + |row| sums -> sigma[b]
//   K3: orthogonal iteration on (Bmat + sigma*I), 512x16 block in LDS.
//       All three matrix ops per iteration use V_WMMA_F32_16X16X4_F32:
//         G = Y^T Y   (A-frag == B-frag trick),  Q = Y * R^{-1}  (CholQR),
//         Y = Bmat @ Q + sigma*Q.
//       Rayleigh quotients -> top-3 eigenpairs -> X = v*sqrt(e)*m.
// ---------------------------------------------------------------------------

#define LDIM 512
#define BDIM 64
#define NCOL 16
#define NITER 48

typedef __attribute__((ext_vector_type(2))) float v2f;
typedef __attribute__((ext_vector_type(8))) float v8f;

__global__ void mds_init_sigma(float* sigma) {
    sigma[threadIdx.x] = 0.0f;
}

// One block (512 threads) per batch: rowsums of masked d2, total S, count n.
__global__ void mds_rowsums(const float* __restrict__ dmap,
                            const int* __restrict__ mask,
                            float* __restrict__ rbuf,
                            float* __restrict__ nS) {
    __shared__ float mloc[LDIM];
    __shared__ float red[LDIM];
    const int b = blockIdx.x;
    const int i = threadIdx.x;

    mloc[i] = (mask[b * LDIM + i] > 0) ? 1.0f : 0.0f;
    __syncthreads();

    const float mi = mloc[i];
    const float* drow = dmap + ((size_t)b * LDIM + i) * LDIM;
    float s = 0.0f;
    for (int j = 0; j < LDIM; ++j) {
        float dd = fminf(fmaxf(drow[j], 0.0f), 100.0f);
        s += (dd * dd + 1e-6f) * mloc[j];
    }
    s *= mi;
    rbuf[b * LDIM + i] = s;

    red[i] = s;
    __syncthreads();
    for (int off = 256; off > 0; off >>= 1) {
        if (i < off) red[i] += red[i + off];
        __syncthreads();
    }
    float S = red[0];
    __syncthreads();
    red[i] = mi;
    __syncthreads();
    for (int off = 256; off > 0; off >>= 1) {
        if (i < off) red[i] += red[i + off];
        __syncthreads();
    }
    if (i == 0) {
        nS[2 * b + 0] = fmaxf(red[0], 1.0f);
        nS[2 * b + 1] = S;
    }
}

// One block (256 threads) per (row i, batch b): write Bmat row + Gershgorin.
__global__ void mds_build_B(const float* __restrict__ dmap,
                            const int* __restrict__ mask,
                            const float* __restrict__ rbuf,
                            const float* __restrict__ nS,
                            float* __restrict__ Bmat,
                            float* __restrict__ sigma) {
    __shared__ float mloc[LDIM];
    __shared__ float rloc[LDIM];
    __shared__ float red[256];
    const int i = blockIdx.x;
    const int b = blockIdx.y;
    const int tid = threadIdx.x;

    for (int j = tid; j < LDIM; j += 256) {
        mloc[j] = (mask[b * LDIM + j] > 0) ? 1.0f : 0.0f;
        rloc[j] = rbuf[b * LDIM + j];
    }
    __syncthreads();

    const float n = nS[2 * b + 0];
    const float S = nS[2 * b + 1];
    const float invn = 1.0f / n;
    const float g = S * invn * invn;
    const float mi = mloc[i];
    const float ri = rloc[i];
    const float* drow = dmap + ((size_t)b * LDIM + i) * LDIM;
    float* Brow = Bmat + ((size_t)b * LDIM + i) * LDIM;

    float asum = 0.0f;
    for (int j = tid; j < LDIM; j += 256) {
        float mj = mloc[j];
        float dd = fminf(fmaxf(drow[j], 0.0f), 100.0f);
        float a = (dd * dd + 1e-6f) * mi * mj;                 // masked d2
        float v = -0.5f * mi * mj * (a - (ri + rloc[j]) * invn + g);
        if (j == i) v += 0.001f * mi;                          // jitter
        Brow[j] = v;
        asum += fabsf(v);
    }
    red[tid] = asum;
    __syncthreads();
    for (int off = 128; off > 0; off >>= 1) {
        if (tid < off) red[tid] += red[tid + off];
        __syncthreads();
    }
    if (tid == 0) {
        // nonneg floats: int compare == float compare; deterministic
        atomicMax(reinterpret_cast<int*>(sigma + b), __float_as_int(red[0]));
    }
}

__device__ __forceinline__ float blk_reduce256(float v, float* red, int tid) {
    red[tid] = v;
    __syncthreads();
    for (int off = 128; off > 0; off >>= 1) {
        if (tid < off) red[tid] += red[tid + off];
        __syncthreads();
    }
    float r = red[0];
    __syncthreads();
    return r;
}

// CholQR: dst (512x16) = orthonormalized src (512x16). All WMMA.
// Gpart: 8*256 floats, Gs: 256, Rinv: 256 (LDS scratch).
__device__ __forceinline__ void cholqr(const float* __restrict__ src,
                                       float* __restrict__ dst,
                                       float* Gpart, float* Gs, float* Rinv,
                                       int tid, int wave, int half, int lm) {
    // --- partial Gram per wave: G_w = src[k0:k0+64]^T src[k0:k0+64] ---
    {
        v8f acc = {0.f, 0.f, 0.f, 0.f, 0.f, 0.f, 0.f, 0.f};
        const int kbeg = wave * (LDIM / 8);
        for (int k0 = kbeg; k0 < kbeg + LDIM / 8; k0 += 4) {
            const int ka = k0 + 2 * half;
            // identical per-lane data serves A (lane=M) and B (lane=N)
            v2f a;
            a.x = src[ka * NCOL + lm];
            a.y = src[(ka + 1) * NCOL + lm];
            acc = __builtin_amdgcn_wmma_f32_16x16x4_f32(
                false, a, false, a, (short)0, acc, false, false);
        }
        for (int r = 0; r < 8; ++r)
            Gpart[wave * 256 + (r + 8 * half) * NCOL + lm] = acc[r];
    }
    __syncthreads();
    // --- reduce 8 partials: one entry per thread ---
    {
        float s = 0.0f;
        for (int w = 0; w < 8; ++w) s += Gpart[w * 256 + tid];
        Gs[tid] = s;
    }
    __syncthreads();
    // --- thread 0: Cholesky G = R^T R and upper-triangular R^{-1} ---
    if (tid == 0) {
        float R[NCOL][NCOL];
        float tr = 0.0f;
        for (int j = 0; j < NCOL; ++j) tr += Gs[j * NCOL + j];
        const float ridge = 1e-10f * tr + 1e-30f;
        for (int j = 0; j < NCOL; ++j) {
            float v = Gs[j * NCOL + j] + ridge;
            for (int k = 0; k < j; ++k) v -= R[k][j] * R[k][j];
            float rjj = sqrtf(fmaxf(v, 1e-20f));
            R[j][j] = rjj;
            float inv = 1.0f / rjj;
            for (int i2 = j + 1; i2 < NCOL; ++i2) {
                float u = Gs[j * NCOL + i2];
                for (int k = 0; k < j; ++k) u -= R[k][j] * R[k][i2];
                R[j][i2] = u * inv;
            }
        }
        for (int c = 0; c < NCOL; ++c) {          // solve R * x = e_c
            float x[NCOL];
            for (int i2 = 0; i2 < NCOL; ++i2) x[i2] = 0.0f;
            x[c] = 1.0f / R[c][c];
            for (int i2 = c - 1; i2 >= 0; --i2) {
                float u = 0.0f;
                for (int k = i2 + 1; k <= c; ++k) u += R[i2][k] * x[k];
                x[i2] = -u / R[i2][i2];
            }
            for (int i2 = 0; i2 < NCOL; ++i2) Rinv[i2 * NCOL + c] = x[i2];
        }
    }
    __syncthreads();
    // --- dst = src @ Rinv : M=512, K=16, N=16 (WMMA) ---
    for (int t = wave; t < LDIM / 16; t += 8) {
        v8f acc = {0.f, 0.f, 0.f, 0.f, 0.f, 0.f, 0.f, 0.f};
        const int rowA = t * 16 + lm;
        for (int k0 = 0; k0 < NCOL; k0 += 4) {
            const int ka = k0 + 2 * half;
            v2f a, bb;
            a.x = src[rowA * NCOL + ka];
            a.y = src[rowA * NCOL + ka + 1];
            bb.x = Rinv[ka * NCOL + lm];
            bb.y = Rinv[(ka + 1) * NCOL + lm];
            acc = __builtin_amdgcn_wmma_f32_16x16x4_f32(
                false, a, false, bb, (short)0, acc, false, false);
        }
        for (int r = 0; r < 8; ++r)
            dst[(t * 16 + r + 8 * half) * NCOL + lm] = acc[r];
    }
    __syncthreads();
}

// One block (256 threads = 8 waves) per batch.
__global__ void __launch_bounds__(256)
mds_subspace_eig(const float* __restrict__ BmatAll,
                 const int* __restrict__ mask,
                 const float* __restrict__ sigmaArr,
                 float* __restrict__ out) {
    __shared__ float Yb[LDIM * NCOL];        // 32 KB
    __shared__ float Qb[LDIM * NCOL];        // 32 KB
    __shared__ float Gpart[8 * NCOL * NCOL]; // 8 KB
    __shared__ float Gs[NCOL * NCOL];
    __shared__ float Rinv[NCOL * NCOL];
    __shared__ float red[256];

    const int b = blockIdx.x;
    const int tid = threadIdx.x;
    const int wave = tid >> 5;
    const int lane = tid & 31;          // wave32
    const int half = lane >> 4;
    const int lm = lane & 15;

    const float* Bm = BmatAll + (size_t)b * LDIM * LDIM;
    const float sigma = sigmaArr[b];

    // init Y with masked DCT-II basis (mutually orthogonal, full rank)
    for (int idx = tid; idx < LDIM * NCOL; idx += 256) {
        int i = idx / NCOL, c = idx % NCOL;
        float mi = (mask[b * LDIM + i] > 0) ? 1.0f : 0.0f;
        Yb[idx] = mi * __cosf(((float)i + 0.5f) * (float)(c + 1) *
                              (3.14159265358979f / (float)LDIM));
    }
    __syncthreads();

    for (int it = 0; it <= NITER; ++it) {
        // Q <- CholQR(Y)   (all-WMMA orthonormalization)
        cholqr(Yb, Qb, Gpart, Gs, Rinv, tid, wave, half, lm);

        // Y = Bmat @ Q + sigma*Q   (32 M-tiles; 8 waves x 4 tiles; K=512)
        for (int t = wave; t < LDIM / 16; t += 8) {
            const int rowA = t * 16 + lm;
            __builtin_prefetch(Bm + (size_t)rowA * LDIM, 0, 1);
            v8f acc = {0.f, 0.f, 0.f, 0.f, 0.f, 0.f, 0.f, 0.f};
            for (int k0 = 0; k0 < LDIM; k0 += 4) {
                const int ka = k0 + 2 * half;
                float2 av = *reinterpret_cast<const float2*>(
                    Bm + (size_t)rowA * LDIM + ka);
                v2f a; a.x = av.x; a.y = av.y;
                v2f bb;
                bb.x = Qb[ka * NCOL + lm];
                bb.y = Qb[(ka + 1) * NCOL + lm];
                acc = __builtin_amdgcn_wmma_f32_16x16x4_f32(
                    false, a, false, bb, (short)0, acc, false, false);
            }
            for (int r = 0; r < 8; ++r) {
                int row = t * 16 + r + 8 * half;
                Yb[row * NCOL + lm] = acc[r] + sigma * Qb[row * NCOL + lm];
            }
        }
        __syncthreads();
    }

    // Qb orthonormal, Yb = (B + sigma*I) Qb -> Rayleigh quotients
    float sq[3];
    for (int k = 0; k < 3; ++k) {
        float part = 0.0f;
        for (int i = tid; i < LDIM; i += 256)
            part += Qb[i * NCOL + k] * Yb[i * NCOL + k];
        float ev = blk_reduce256(part, red, tid) - sigma;
        sq[k] = sqrtf(fmaxf(ev, 0.0f));
    }
    // eigh is ascending: out col 0 = 3rd-largest ... col 2 = largest
    for (int idx = tid; idx < LDIM * 3; idx += 256) {
        int i = idx / 3, c = idx % 3;
        int k = 2 - c;
        float mi = (mask[b * LDIM + i] > 0) ? 1.0f : 0.0f;
        out[(size_t)b * LDIM * 3 + idx] = Qb[i * NCOL + k] * sq[k] * mi;
    }
}

extern "C" void kernel_launch(void* const* d_in, const int* in_sizes, int n_in,
                              void* d_out, int out_size, void* d_ws, size_t ws_size,
                              hipStream_t stream) {
    const float* dmap = (const float*)d_in[0];   // (64,512,512) f32
    const int*   mask = (const int*)d_in[1];     // (64,512) i32
    float* out = (float*)d_out;                  // (64,512,3) f32

    float* ws   = (float*)d_ws;
    float* Bmat = ws;                                      // 64*512*512 floats
    float* rbuf = ws + (size_t)BDIM * LDIM * LDIM;         // 64*512
    float* nS   = rbuf + (size_t)BDIM * LDIM;              // 128
    float* sig  = nS + 2 * BDIM;                           // 64

    mds_init_sigma<<<1, BDIM, 0, stream>>>(sig);
    mds_rowsums<<<BDIM, LDIM, 0, stream>>>(dmap, mask, rbuf, nS);
    mds_build_B<<<dim3(LDIM, BDIM), 256, 0, stream>>>(dmap, mask, rbuf, nS,
                                                      Bmat, sig);
    mds_subspace_eig<<<BDIM, 256, 0, stream>>>(Bmat, mask, sig, out);
}